// SetAbstraction_84052509982737
// MI455X (gfx1250) — compile-verified
//
#include <hip/hip_runtime.h>
#include <hip/hip_bf16.h>

typedef __attribute__((ext_vector_type(2))) float v2f;
typedef __attribute__((ext_vector_type(8))) float v8f;

#define NPTS 262144
#define MS   65536
#define KNN  32
#define CIN  64
#define CK   72           // 3 + 64 = 67, padded to 72 (multiple of 8 for 2x-unrolled k=4 WMMA steps)
#define KREAL 67
#define COUT 128
#define BN_EPS 1e-5f

// ---------------------------------------------------------------------------
// Kernel 0: zero the BN accumulators, emit n_o = [M]
// ---------------------------------------------------------------------------
__global__ void k_init(float* __restrict__ stats, int* __restrict__ n_o_slot) {
    int t = threadIdx.x;
    if (t < 2 * COUT) stats[t] = 0.0f;
    if (t == 0) *n_o_slot = MS;
}

// ---------------------------------------------------------------------------
// Kernel 1: gather K neighbors, concat rel-xyz + features, max-pool over K.
// One wave32 per sample. Lane c owns channels c and c+32 of x; lanes 0..2
// additionally own the rel-xyz components. knn indices preloaded coalesced
// and broadcast with __shfl; next neighbor row prefetched to hide latency.
// ---------------------------------------------------------------------------
__global__ void k_gather_pool(const float* __restrict__ p,
                              const float* __restrict__ x,
                              const int*   __restrict__ idx,
                              const int*   __restrict__ knn,
                              float* __restrict__ pooled,
                              float* __restrict__ n_p) {
    const int lane = threadIdx.x & 31;
    const int wave = threadIdx.x >> 5;
    const int m    = blockIdx.x * 8 + wave;            // blockDim = 256 -> 8 samples/block

    const int nk = knn[(size_t)m * KNN + lane];        // coalesced preload of all 32 indices
    const int s  = idx[m];

    float cv = 0.0f;
    if (lane < 3) {
        cv = p[(size_t)s * 3 + lane];
        n_p[(size_t)m * 3 + lane] = cv;                // sampler output n_p = p[idx]
    }

    float mx0  = -3.402823466e38f;
    float mx1  = -3.402823466e38f;
    float rmax = -3.402823466e38f;

    #pragma unroll 4
    for (int k = 0; k < KNN; ++k) {
        const int nbr = __shfl(nk, k, 32);
        if (k + 1 < KNN) {
            const int nxt = __shfl(nk, k + 1, 32);
            __builtin_prefetch(&x[(size_t)nxt * CIN + lane], 0, 3);
        }
        const float a0 = x[(size_t)nbr * CIN + lane];        // 128B coalesced
        const float a1 = x[(size_t)nbr * CIN + 32 + lane];   // 128B coalesced
        mx0 = fmaxf(mx0, a0);
        mx1 = fmaxf(mx1, a1);
        if (lane < 3) {
            const float rv = p[(size_t)nbr * 3 + lane] - cv;
            rmax = fmaxf(rmax, rv);
        }
    }

    float* prow = pooled + (size_t)m * CK;
    prow[3 + lane]  = mx0;                              // channels 3..34
    prow[35 + lane] = mx1;                              // channels 35..66
    if (lane < 3) prow[lane] = rmax;                    // rel-xyz 0..2
    if (lane >= 3 && lane < 8) prow[64 + lane] = 0.0f;  // zero pad cols 67..71
}

// ---------------------------------------------------------------------------
// Kernel 2: h = pooled[M,72] @ Wpad[72,128] + b  via V_WMMA_F32_16X16X4_F32.
// Block = 256 threads (8 waves); block owns a 16-row stripe, wave w owns
// 16 columns [16w, 16w+16). 18 k-steps of 4, split over two accumulators so
// consecutive WMMAs are independent (hides D->C RAW hazard NOPs).
// ---------------------------------------------------------------------------
__global__ void k_gemm_wmma(const float* __restrict__ pooled,
                            const float* __restrict__ W,     // [67,128] row-major
                            const float* __restrict__ bias,
                            float* __restrict__ h) {         // [M,128]
    const int lane = threadIdx.x & 31;
    const int wave = threadIdx.x >> 5;
    const int m0   = blockIdx.x * 16;
    const int n0   = wave * 16;
    const int r    = lane & 15;          // row/col within tile
    const int hi   = lane >> 4;          // 0: K pair {0,1} / rows 0-7 ; 1: K pair {2,3} / rows 8-15
    const int col  = n0 + r;

    const float* arow = pooled + (size_t)(m0 + r) * CK + 2 * hi;

    v8f acc0 = {};
    v8f acc1 = {};

    #pragma unroll
    for (int k0 = 0; k0 < CK; k0 += 8) {
        // even k-step (k0 .. k0+3)
        {
            const v2f a = *(const v2f*)(arow + k0);               // 8B-aligned: CK even, offsets even
            const int ka = k0 + 2 * hi;
            v2f b;
            b.x = (ka     < KREAL) ? W[(size_t)ka * COUT + col]       : 0.0f;
            b.y = (ka + 1 < KREAL) ? W[(size_t)(ka + 1) * COUT + col] : 0.0f;
            acc0 = __builtin_amdgcn_wmma_f32_16x16x4_f32(
                false, a, false, b, (short)0, acc0, false, false);
        }
        // odd k-step (k0+4 .. k0+7) -> independent accumulator
        {
            const v2f a = *(const v2f*)(arow + k0 + 4);
            const int kb = k0 + 4 + 2 * hi;
            v2f b;
            b.x = (kb     < KREAL) ? W[(size_t)kb * COUT + col]       : 0.0f;
            b.y = (kb + 1 < KREAL) ? W[(size_t)(kb + 1) * COUT + col] : 0.0f;
            acc1 = __builtin_amdgcn_wmma_f32_16x16x4_f32(
                false, a, false, b, (short)0, acc1, false, false);
        }
    }

    const float bv = bias[col];
    #pragma unroll
    for (int v = 0; v < 8; ++v) {
        const int row = m0 + v + hi * 8;                 // C/D VGPR layout
        h[(size_t)row * COUT + col] = (acc0[v] + acc1[v]) + bv;
    }
}

// ---------------------------------------------------------------------------
// Kernel 3: per-channel sum / sum-of-squares (blockDim = 128, thread = column)
// ---------------------------------------------------------------------------
__global__ void k_stats(const float* __restrict__ h, float* __restrict__ stats) {
    const int col = threadIdx.x;
    float s = 0.0f, s2 = 0.0f;
    for (int m = blockIdx.x; m < MS; m += gridDim.x) {
        const float v = h[(size_t)m * COUT + col];       // coalesced across threads
        s += v; s2 += v * v;
    }
    atomicAdd(&stats[col], s);
    atomicAdd(&stats[COUT + col], s2);
}

// ---------------------------------------------------------------------------
// Kernel 4: BatchNorm (batch stats) + ReLU, in place. Each thread sticks to
// one column so scale/shift (incl. rsqrt) are computed once.
// ---------------------------------------------------------------------------
__global__ void k_bn_relu(float* __restrict__ h,
                          const float* __restrict__ stats,
                          const float* __restrict__ gamma,
                          const float* __restrict__ beta) {
    const int t   = blockIdx.x * blockDim.x + threadIdx.x;
    const int col = t & (COUT - 1);
    const int r0  = t >> 7;
    const int rstride = (gridDim.x * blockDim.x) >> 7;

    const float inv_m = 1.0f / (float)MS;
    const float mean  = stats[col] * inv_m;
    const float var   = stats[COUT + col] * inv_m - mean * mean;
    const float rstd  = rsqrtf(var + BN_EPS);
    const float sc    = rstd * gamma[col];
    const float sh    = beta[col] - mean * sc;

    for (int m = r0; m < MS; m += rstride) {
        float v = h[(size_t)m * COUT + col];
        v = v * sc + sh;
        h[(size_t)m * COUT + col] = fmaxf(v, 0.0f);
    }
}

// ---------------------------------------------------------------------------
// Launcher
// inputs: 0=p[N,3] 1=x[N,64] 2=o[1] 3=idx[M] 4=knn_idx[M,32]
//         5=W[67,128] 6=b[128] 7=gamma[128] 8=beta[128]
// d_out:  n_p[M*3] | x_out[M*128] | n_o[1]
// ---------------------------------------------------------------------------
extern "C" void kernel_launch(void* const* d_in, const int* in_sizes, int n_in,
                              void* d_out, int out_size, void* d_ws, size_t ws_size,
                              hipStream_t stream) {
    const float* p     = (const float*)d_in[0];
    const float* x     = (const float*)d_in[1];
    const int*   idx   = (const int*)d_in[3];
    const int*   knn   = (const int*)d_in[4];
    const float* W     = (const float*)d_in[5];
    const float* b     = (const float*)d_in[6];
    const float* gamma = (const float*)d_in[7];
    const float* beta  = (const float*)d_in[8];

    float* out   = (float*)d_out;
    float* n_p   = out;                      // [M,3]
    float* x_out = out + (size_t)3 * MS;     // [M,128] (also GEMM output buffer h)
    int*   n_o   = (int*)(out + (size_t)131 * MS);

    float* pooled = (float*)d_ws;                        // [M,72] = 18.9 MB
    float* stats  = pooled + (size_t)MS * CK;            // [2,128]

    k_init<<<1, 256, 0, stream>>>(stats, n_o);
    k_gather_pool<<<MS / 8, 256, 0, stream>>>(p, x, idx, knn, pooled, n_p);
    k_gemm_wmma<<<MS / 16, 256, 0, stream>>>(pooled, W, b, x_out);
    k_stats<<<512, COUT, 0, stream>>>(x_out, stats);
    k_bn_relu<<<1024, 256, 0, stream>>>(x_out, stats, gamma, beta);
}